// PQCGuided_90546500534603
// MI455X (gfx1250) — compile-verified
//
#include <hip/hip_runtime.h>

typedef float v2f __attribute__((ext_vector_type(2)));
typedef float v8f __attribute__((ext_vector_type(8)));

#define NUM_WIRES 11
#define NUM_LAYERS 8
#define DIM 2048      // 2^11 amplitudes
#define NPIX 1024
#define BATCH 1024
#define GSTRIDE (NUM_LAYERS * 3 * 256)   // per re/im plane of fused gates

// ---------------------------------------------------------------------------
// Prep kernel: build the three fused 16x16 complex unitaries per layer.
//   wr = pi * tanh(w);  Rot(phi,theta,omega) per (layer,wire);
//   G0 = U0(x)U1(x)U2(x)U3, G1 = U4(x)U5(x)U6(x)U7,
//   G2 = blockdiag( U8(x)U9(x)U10 , same )  (acts on low 3 bits, bit3 = id)
// Stored row-major: gmat[(l*3+g)*256 + r*16 + c]  (re plane), +GSTRIDE (im).
// ---------------------------------------------------------------------------
__global__ void __launch_bounds__(256)
pqc_prep(const float* __restrict__ weights, float* __restrict__ gmat) {
    __shared__ float Ure[NUM_LAYERS][NUM_WIRES][2][2];
    __shared__ float Uim[NUM_LAYERS][NUM_WIRES][2][2];
    const int tid = threadIdx.x;
    if (tid < NUM_LAYERS * NUM_WIRES) {
        const int l = tid / NUM_WIRES, q = tid % NUM_WIRES;
        const float PI = 3.14159265358979323846f;
        const float phi   = PI * tanhf(weights[(l * NUM_WIRES + q) * 3 + 0]);
        const float theta = PI * tanhf(weights[(l * NUM_WIRES + q) * 3 + 1]);
        const float omega = PI * tanhf(weights[(l * NUM_WIRES + q) * 3 + 2]);
        const float c = cosf(0.5f * theta), s = sinf(0.5f * theta);
        const float a = 0.5f * (phi + omega), b = 0.5f * (phi - omega);
        const float ca = cosf(a), sa = sinf(a), cb = cosf(b), sb = sinf(b);
        // Rot = [[ep*c, -em*s], [conj(em)*s, conj(ep)*c]],
        // ep = ca - i sa, em = cb + i sb
        Ure[l][q][0][0] =  ca * c;  Uim[l][q][0][0] = -sa * c;
        Ure[l][q][0][1] = -cb * s;  Uim[l][q][0][1] = -sb * s;
        Ure[l][q][1][0] =  cb * s;  Uim[l][q][1][0] = -sb * s;
        Ure[l][q][1][1] =  ca * c;  Uim[l][q][1][1] =  sa * c;
    }
    __syncthreads();

    const int r  = tid >> 4;     // 0..15
    const int cc = tid & 15;     // 0..15
    for (int l = 0; l < NUM_LAYERS; ++l) {
        for (int g = 0; g < 3; ++g) {
            float re, im;
            if (g < 2) {
                const int w0 = g * 4;          // wires w0..w0+3, w0 is MSB
                re = 1.f; im = 0.f;
#pragma unroll
                for (int w = 0; w < 4; ++w) {
                    const int rb  = (r  >> (3 - w)) & 1;
                    const int cb2 = (cc >> (3 - w)) & 1;
                    const float ur = Ure[l][w0 + w][rb][cb2];
                    const float ui = Uim[l][w0 + w][rb][cb2];
                    const float nr = re * ur - im * ui;
                    const float ni = re * ui + im * ur;
                    re = nr; im = ni;
                }
            } else {
                if ((r >> 3) != (cc >> 3)) { re = 0.f; im = 0.f; }
                else {
                    re = 1.f; im = 0.f;
                    const int r3 = r & 7, c3 = cc & 7;   // wires 8..10, wire8 MSB
#pragma unroll
                    for (int w = 0; w < 3; ++w) {
                        const int rb  = (r3 >> (2 - w)) & 1;
                        const int cb2 = (c3 >> (2 - w)) & 1;
                        const float ur = Ure[l][8 + w][rb][cb2];
                        const float ui = Uim[l][8 + w][rb][cb2];
                        const float nr = re * ur - im * ui;
                        const float ni = re * ui + im * ur;
                        re = nr; im = ni;
                    }
                }
            }
            const int base = (l * 3 + g) * 256 + r * 16 + cc;
            gmat[base]           = re;
            gmat[GSTRIDE + base] = im;
        }
    }
}

// Flat state index for (k = contracted 4-bit group value, n = column 0..127)
//   G=0: k = bits 10..7 (wires 0-3)
//   G=1: k = bits  6..3 (wires 4-7), n = {bits10..7, bits2..0}
//   G=2: k = bits  3..0 (wire7 + wires 8-10)
template <int G>
__device__ __forceinline__ int map_idx(int k, int n) {
    if constexpr (G == 0) return (k << 7) + n;
    else if constexpr (G == 1) return ((n >> 3) << 7) + (k << 3) + (n & 7);
    else return (n << 4) + k;
}

// Apply one fused 16x16 complex unitary (group G) to the LDS state via
// V_WMMA_F32_16X16X4_F32. Fully unrolled: 32 straight-line WMMAs per call.
template <int G>
__device__ __forceinline__ void apply_group(const float* __restrict__ Gre,
                                            const float* __restrict__ Gim,
                                            float* __restrict__ sRe,
                                            float* __restrict__ sIm,
                                            int gb, int lane, int wave) {
    v8f accRe[2] = {};
    v8f accIm[2] = {};
    const int m  = lane & 15;
    const int kh = (lane >> 4) << 1;         // lane-half K offset: 0 or 2
#pragma unroll
    for (int ks = 0; ks < 4; ++ks) {
        const int kb = ks * 4 + kh;
        // A fragment (16x4 slab of G): lane row m, K = kb, kb+1
        v2f are, aim, ain;
        are.x = Gre[gb + m * 16 + kb];  are.y = Gre[gb + m * 16 + kb + 1];
        aim.x = Gim[gb + m * 16 + kb];  aim.y = Gim[gb + m * 16 + kb + 1];
        ain.x = -aim.x;                 ain.y = -aim.y;
#pragma unroll
        for (int tt = 0; tt < 2; ++tt) {
            const int n  = (wave * 2 + tt) * 16 + m;   // column
            const int i0 = map_idx<G>(kb,     n);
            const int i1 = map_idx<G>(kb + 1, n);
            v2f bre, bim;
            bre.x = sRe[i0];  bre.y = sRe[i1];
            bim.x = sIm[i0];  bim.y = sIm[i1];
            // D_re += Ar*Br - Ai*Bi ; D_im += Ar*Bi + Ai*Br
            accRe[tt] = __builtin_amdgcn_wmma_f32_16x16x4_f32(
                false, are, false, bre, (short)0, accRe[tt], false, false);
            accRe[tt] = __builtin_amdgcn_wmma_f32_16x16x4_f32(
                false, ain, false, bim, (short)0, accRe[tt], false, false);
            accIm[tt] = __builtin_amdgcn_wmma_f32_16x16x4_f32(
                false, are, false, bim, (short)0, accIm[tt], false, false);
            accIm[tt] = __builtin_amdgcn_wmma_f32_16x16x4_f32(
                false, aim, false, bre, (short)0, accIm[tt], false, false);
        }
    }
    __syncthreads();   // all B reads done before overwrite
    const int mh = (lane >> 4) << 3;         // C/D: lanes 16-31 hold M+8
#pragma unroll
    for (int tt = 0; tt < 2; ++tt) {
        const int n = (wave * 2 + tt) * 16 + m;
#pragma unroll
        for (int v = 0; v < 8; ++v) {
            const int ii = map_idx<G>(v + mh, n);
            sRe[ii] = accRe[tt][v];
            sIm[ii] = accIm[tt][v];
        }
    }
    __syncthreads();
}

// ---------------------------------------------------------------------------
// Main kernel: one block (4 wave32) per batch element; state in LDS.
// ---------------------------------------------------------------------------
__global__ void __launch_bounds__(128)
pqc_main(const float* __restrict__ image, const float* __restrict__ label,
         const float* __restrict__ gmat, float* __restrict__ out) {
    __shared__ float sRe[DIM];
    __shared__ float sIm[DIM];
    __shared__ float red[128];

    const int b    = blockIdx.x;
    const int tid  = threadIdx.x;
    const int lane = tid & 31;          // wave32
    const int wave = tid >> 5;

    // --- normalize image row, build initial state with RX(label) on wire 10 ---
    float x[8];
    float ss = 0.f;
#pragma unroll
    for (int e = 0; e < 8; ++e) {
        x[e] = image[b * NPIX + tid + 128 * e];
        ss += x[e] * x[e];
    }
    red[tid] = ss;
    __syncthreads();
    for (int off = 64; off > 0; off >>= 1) {
        if (tid < off) red[tid] += red[tid + off];
        __syncthreads();
    }
    const float inv = 1.0f / sqrtf(red[0]);
    const float lb  = label[b];
    const float cl  = cosf(0.5f * lb);
    const float sl  = -sinf(0.5f * lb);   // amplitude -i*sin -> imag part
#pragma unroll
    for (int e = 0; e < 8; ++e) {
        const int k = tid + 128 * e;
        const float v = x[e] * inv;
        sRe[2 * k]     = cl * v;  sIm[2 * k]     = 0.f;
        sRe[2 * k + 1] = 0.f;     sIm[2 * k + 1] = sl * v;
    }
    __syncthreads();

    const float* __restrict__ Gre = gmat;
    const float* __restrict__ Gim = gmat + GSTRIDE;

    for (int l = 0; l < NUM_LAYERS; ++l) {
        // ---- fused 16x16 complex unitaries via V_WMMA_F32_16X16X4_F32 ----
        const int gb = l * 3 * 256;
        apply_group<0>(Gre, Gim, sRe, sIm, gb,           lane, wave);
        apply_group<1>(Gre, Gim, sRe, sIm, gb + 256,     lane, wave);
        apply_group<2>(Gre, Gim, sRe, sIm, gb + 512,     lane, wave);

        // ---- CNOT ring: GF(2)-linear index permutation, LDS shuffle ----
        const int rsh = l % (NUM_WIRES - 1) + 1;
        float pr[16], pim[16];
        int   pj[16];
#pragma unroll
        for (int e = 0; e < 16; ++e) {
            const int i = tid + 128 * e;
            pr[e]  = sRe[i];
            pim[e] = sIm[i];
            int j = i;
#pragma unroll
            for (int q = 0; q < NUM_WIRES; ++q) {
                int t = q + rsh; if (t >= NUM_WIRES) t -= NUM_WIRES;
                const int cpos = 10 - q;             // wire q -> bit 10-q
                const int tpos = 10 - t;
                j ^= ((j >> cpos) & 1) << tpos;      // target ^= control
            }
            pj[e] = j;
        }
        __syncthreads();
#pragma unroll
        for (int e = 0; e < 16; ++e) {
            sRe[pj[e]] = pr[e];
            sIm[pj[e]] = pim[e];
        }
        __syncthreads();
    }

    // ---- probabilities of even amplitudes (wire10 == 0), * NUM_PIXELS ----
#pragma unroll
    for (int e = 0; e < 8; ++e) {
        const int k = tid + 128 * e;
        const float re = sRe[2 * k], im = sIm[2 * k];
        out[b * NPIX + k] = (re * re + im * im) * (float)NPIX;
    }
}

extern "C" void kernel_launch(void* const* d_in, const int* in_sizes, int n_in,
                              void* d_out, int out_size, void* d_ws, size_t ws_size,
                              hipStream_t stream) {
    const float* image   = (const float*)d_in[0];   // (1024,1,32,32) f32
    const float* label   = (const float*)d_in[1];   // (1024,) f32
    const float* weights = (const float*)d_in[2];   // (8,11,3) f32
    float* gmat = (float*)d_ws;                     // 2*8*3*256 floats = 48 KB
    float* outp = (float*)d_out;                    // (1024,1,32,32) f32

    pqc_prep<<<1, 256, 0, stream>>>(weights, gmat);
    pqc_main<<<BATCH, 128, 0, stream>>>(image, label, gmat, outp);
}